// CGLContrastiveLoss_11312943857885
// MI455X (gfx1250) — compile-verified
//
#include <hip/hip_runtime.h>
#include <math.h>

typedef __attribute__((ext_vector_type(2))) float v2f;
typedef __attribute__((ext_vector_type(8))) float v8f;

#define NREG        25
#define RDIM        1024
#define ROWF        1028                 // padded floats per region row (bank-conflict-free WMMA fragment loads)
#define NROWS       32                   // padded region rows per augmentation (rows 25..31 zero)
#define GRAM_LD     33
#define REG_FLOATS  (2 * NROWS * ROWF)   // 65792
#define GRAM_OFF    REG_FLOATS
#define GRAM_FLOATS (3 * 32 * GRAM_LD)   // 3168
#define RED_OFF     (GRAM_OFF + GRAM_FLOATS)
#define NORM_OFF    (RED_OFF + 32)
#define SMEM_FLOATS (NORM_OFF + 64)      // ~276 KB of the 320 KB WGP LDS
#define THREADS     384                  // 12 wave32 waves
#define TAU_INV     10.0f

__global__ void cgl_batch_kernel(const float* __restrict__ proj,
                                 float* __restrict__ partials, int B)
{
    extern __shared__ float smem[];
    const int tid  = threadIdx.x;
    const int lane = tid & 31;
    const int wid  = tid >> 5;
    const int b    = blockIdx.x;

    // ---- zero padded region storage (rows >= 25 must be 0 for WMMA tiles) ----
    for (int i = tid; i < REG_FLOATS; i += THREADS) smem[i] = 0.0f;
    __syncthreads();

    // ---- gather: regions are (c, dy, dx) windows at centres 10..50 step 10 ----
    for (int idx = tid; idx < 2 * NREG * RDIM; idx += THREADS) {
        int a   = idx / (NREG * RDIM);
        int rem = idx - a * (NREG * RDIM);
        int r   = rem >> 10;
        int d   = rem & 1023;
        int c   = d >> 2;
        int dy  = (d >> 1) & 1;
        int dx  = d & 1;
        int h   = 10 * (r / 5 + 1) - 1 + dy;   // centres[i] = 10*(i+1); offs = {-1, 0}
        int w   = 10 * (r % 5 + 1) - 1 + dx;
        long img = (long)((a == 0) ? (B + b) : (2 * B + b));
        long g   = (img * 256 + c) * 4096 + (long)h * 64 + w;
        smem[(a * NROWS + r) * ROWF + d] = proj[g];
    }
    __syncthreads();

    // ---- per-region L2 norms (wave32 shuffle reduction) ----
    for (int j = wid; j < 2 * NREG; j += THREADS / 32) {
        int a = j / NREG, r = j - a * NREG;
        const float* base = smem + (a * NROWS + r) * ROWF;
        float acc = 0.0f;
        for (int k = lane; k < RDIM; k += 32) { float v = base[k]; acc += v * v; }
        for (int off = 16; off > 0; off >>= 1) acc += __shfl_xor(acc, off, 32);
        if (lane == 0) smem[NORM_OFF + j] = 1.0f / fmaxf(sqrtf(acc), 1e-12f);
    }
    __syncthreads();

    // ---- scale in place ----
    for (int idx = tid; idx < 2 * NREG * RDIM; idx += THREADS) {
        int j = idx >> 10;
        int d = idx & 1023;
        int a = j / NREG, r = j - a * NREG;
        smem[(a * NROWS + r) * ROWF + d] *= smem[NORM_OFF + j];
    }
    __syncthreads();

    // ---- Gram matrices via V_WMMA_F32_16X16X4_F32 -----------------------------
    // 12 waves = 3 matrices (S11, S22, S12) x 4 (16x16) tile pairs; K = 1024.
    {
        int m  = wid >> 2;                 // 0: S11, 1: S22, 2: S12
        int p  = wid & 3;
        int ti = p >> 1, tj = p & 1;
        int ma = (m == 1) ? 1 : 0;
        int mb = (m == 0) ? 0 : 1;
        int ro = lane & 15;                // row within tile (A) / col (B)
        int ko = (lane >> 4) << 1;         // K sub-offset: lanes 0-15 -> K,K+1; 16-31 -> K+2,K+3
        const float* aBase = smem + (ma * NROWS + ti * 16 + ro) * ROWF + ko;
        const float* bBase = smem + (mb * NROWS + tj * 16 + ro) * ROWF + ko;
        v8f acc = {0.f, 0.f, 0.f, 0.f, 0.f, 0.f, 0.f, 0.f};
        for (int k = 0; k < RDIM; k += 4) {
            v2f av = *(const v2f*)(aBase + k);
            v2f bv = *(const v2f*)(bBase + k);
            acc = __builtin_amdgcn_wmma_f32_16x16x4_f32(
                false, av, false, bv, (short)0, acc, false, false);
        }
        // D layout: VGPR v -> row v + 8*(lane>=16), col = lane & 15
        int colg  = tj * 16 + ro;
        int rbase = ti * 16 + ((lane >> 4) << 3);
#pragma unroll
        for (int v = 0; v < 8; ++v)
            smem[GRAM_OFF + (m * 32 + rbase + v) * GRAM_LD + colg] = acc[v];
    }
    __syncthreads();

    // ---- masked logsumexp; pos = diag(S12)/tau --------------------------------
    if (tid < NREG) {
        int r = tid;
        const float* g11 = smem + GRAM_OFF + 0 * 32 * GRAM_LD;
        const float* g22 = smem + GRAM_OFF + 1 * 32 * GRAM_LD;
        const float* g12 = smem + GRAM_OFF + 2 * 32 * GRAM_LD;
        float pos = g12[r * GRAM_LD + r] * TAU_INV;
        float m1 = pos, m2 = pos;
        for (int s = 0; s < NREG; ++s) {
            if (s == r) continue;
            m1 = fmaxf(m1, fmaxf(g11[r * GRAM_LD + s] * TAU_INV,
                                 g12[r * GRAM_LD + s] * TAU_INV));
            m2 = fmaxf(m2, fmaxf(g22[r * GRAM_LD + s] * TAU_INV,
                                 g12[s * GRAM_LD + r] * TAU_INV));
        }
        float sum1 = expf(pos - m1), sum2 = expf(pos - m2);
        for (int s = 0; s < NREG; ++s) {
            if (s == r) continue;
            sum1 += expf(g11[r * GRAM_LD + s] * TAU_INV - m1)
                  + expf(g12[r * GRAM_LD + s] * TAU_INV - m1);
            sum2 += expf(g22[r * GRAM_LD + s] * TAU_INV - m2)
                  + expf(g12[s * GRAM_LD + r] * TAU_INV - m2);
        }
        float lse1 = m1 + logf(sum1);
        float lse2 = m2 + logf(sum2);
        smem[RED_OFF + r] = (lse1 - pos) + (lse2 - pos);
    }
    __syncthreads();

    if (tid == 0) {
        float t = 0.0f;
        for (int r = 0; r < NREG; ++r) t += smem[RED_OFF + r];
        partials[b] = t;
    }
}

__global__ void cgl_finalize_kernel(const float* __restrict__ partials,
                                    float* __restrict__ out, int B)
{
    if (threadIdx.x == 0 && blockIdx.x == 0) {
        float t = 0.0f;
        for (int b = 0; b < B; ++b) t += partials[b];   // fixed order -> deterministic
        out[0] = t * (0.5f / (float)NREG / (float)B);   // loss/2/R/B * LOCAL_ALPHA
    }
}

extern "C" void kernel_launch(void* const* d_in, const int* in_sizes, int n_in,
                              void* d_out, int out_size, void* d_ws, size_t ws_size,
                              hipStream_t stream)
{
    (void)n_in; (void)out_size; (void)ws_size;
    const float* proj = (const float*)d_in[0];
    int B = in_sizes[0] / (3 * 256 * 64 * 64);   // = 32 for the reference shapes
    float* partials = (float*)d_ws;              // B floats of scratch

    size_t smem_bytes = (size_t)SMEM_FLOATS * sizeof(float);
    cgl_batch_kernel<<<B, THREADS, smem_bytes, stream>>>(proj, partials, B);
    cgl_finalize_kernel<<<1, 32, 0, stream>>>(partials, (float*)d_out, B);
}